// PredictorCorrectorRFM2D_84250078478886
// MI455X (gfx1250) — compile-verified
//
#include <hip/hip_runtime.h>
#include <math.h>

// ---------------- problem constants ----------------
#define NN   128            // grid
#define MM   128            // features
#define BB   8              // batch
#define NP   16384          // N*N
#define PIT  130            // LDS row pitch (floats) to avoid bank conflicts
#define INVH 127.0f
#define C1_SMOOTH (0.03f * 0.0001f * 127.0f * 127.0f)   // DT*ETA/H^2
#define PI_F 3.14159265358979323846f

typedef float v2f __attribute__((ext_vector_type(2)));
typedef float v8f __attribute__((ext_vector_type(8)));

// ---------------- workspace layout (floats) ----------------
#define OFF_AINV   0
#define OFF_DLX    (OFF_AINV + BB*NP)
#define OFF_DLY    (OFF_DLX  + BB*NP)
#define OFF_ACCX   (OFF_DLY  + BB*NP)
#define OFF_ACCY   (OFF_ACCX + BB*NP)
#define OFF_RHS1   (OFF_ACCY + BB*NP)
#define OFF_SALPHA (OFF_RHS1 + BB*NP)
#define OFF_SMAT   (OFF_SALPHA + NP)
#define OFF_DINV   (OFF_SMAT + NP)
#define OFF_S1     (OFF_DINV + NP)
#define OFF_S2     (OFF_S1 + MM*NP)

// =====================================================================
// 128x128x128 fp32 GEMM on LDS tiles using V_WMMA_F32_16X16X4_F32.
// 256 threads = 8 waves; wave w owns column-strip ct=w (8 tiles of 16x16).
// MODE 0: D->ldsD          MODE 1: D*DinvG -> ldsD
// MODE 2: D -> ldsD shifted (+1,+1)  (builds zero-padded p0)
// MODE 3: D -> global out shifted (+1,+1)
// =====================================================================
template<int MODE>
__device__ __forceinline__ void gemm128(const float* aL, const float* bL,
                                        float* dL, const float* scaleG,
                                        float* outG)
{
    const int tid  = threadIdx.x;
    const int wave = tid >> 5;
    const int lane = tid & 31;
    const int half = lane >> 4;        // 0: K=0,1   1: K=2,3
    const int l16  = lane & 15;
    const int ct   = wave;             // column tile
    const int col  = ct * 16 + l16;

    v8f acc[8];
    const v8f vzero = {0.f,0.f,0.f,0.f,0.f,0.f,0.f,0.f};
#pragma unroll
    for (int r = 0; r < 8; ++r) acc[r] = vzero;

    for (int k = 0; k < 128; k += 4) {
        const int kk = k + 2 * half;
        v2f bfrag;
        bfrag.x = bL[kk * PIT + col];
        bfrag.y = bL[(kk + 1) * PIT + col];
#pragma unroll
        for (int rt = 0; rt < 8; ++rt) {
            const int row = rt * 16 + l16;
            v2f afrag;
            afrag.x = aL[row * PIT + kk];
            afrag.y = aL[row * PIT + kk + 1];
            acc[rt] = __builtin_amdgcn_wmma_f32_16x16x4_f32(
                false, afrag, false, bfrag, (short)0, acc[rt], false, false);
        }
    }

#pragma unroll
    for (int rt = 0; rt < 8; ++rt) {
#pragma unroll
        for (int i = 0; i < 8; ++i) {
            const int m = rt * 16 + i + 8 * half;
            const int n = col;
            float v = acc[rt][i];
            if (MODE == 0) {
                dL[m * PIT + n] = v;
            } else if (MODE == 1) {
                dL[m * PIT + n] = v * scaleG[m * NN + n];
            } else if (MODE == 2) {
                if (m < 127 && n < 127) dL[(m + 1) * PIT + (n + 1)] = v;
            } else {
                if (m < 127 && n < 127) outG[(m + 1) * NN + (n + 1)] = v;
            }
        }
    }
}

// =====================================================================
// init: DST matrix S, eigenvalue scale Dinv, zero atomic accumulators
// =====================================================================
__global__ void init_kernel(float* ws)
{
    int i = blockIdx.x * blockDim.x + threadIdx.x;   // 262144 threads
    if (i < NP) {
        int r = i >> 7, c = i & 127;
        float s = 0.f, d = 0.f;
        if (r < 126 && c < 126) {
            int phase = ((r + 1) * (c + 1)) % 254;   // exact arg reduction
            s = __sinf((float)phase * (PI_F / 127.0f));
            float l1 = 2.0f - 2.0f * __cosf(PI_F * (float)(r + 1) / 127.0f);
            float l2 = 2.0f - 2.0f * __cosf(PI_F * (float)(c + 1) / 127.0f);
            d = 4.0f / (260144641.0f * (l1 + l2));   // 4 / (127^4 * (l1+l2))
        }
        ws[OFF_SMAT + i] = s;
        ws[OFF_DINV + i] = d;
    }
    if (i < 2 * BB * NP) ws[OFF_ACCX + i] = 0.f;     // accx+accy contiguous
}

// =====================================================================
// smooth_coefficient (34 diffusion steps) + a_inv + grad(log a)
// one block per b; whole field in LDS (double buffered)
// =====================================================================
__global__ void smooth_kernel(const float* __restrict__ a, float* ws)
{
    extern __shared__ float sm[];
    float* u0 = sm;
    float* u1 = sm + NP;
    const int b = blockIdx.x;
    const float* ab = a + b * NP;

    for (int p = threadIdx.x; p < NP; p += 256) u0[p] = ab[p];
    __syncthreads();

    for (int s = 0; s < 34; ++s) {
        float* src = (s & 1) ? u1 : u0;
        float* dst = (s & 1) ? u0 : u1;
        for (int p = threadIdx.x; p < NP; p += 256) {
            int x = p >> 7, y = p & 127;
            float c = src[p];
            float up = src[(x > 0 ? x - 1 : 0) * NN + y];
            float dn = src[(x < 127 ? x + 1 : 127) * NN + y];
            float lf = src[x * NN + (y > 0 ? y - 1 : 0)];
            float rt = src[x * NN + (y < 127 ? y + 1 : 127)];
            dst[p] = c + C1_SMOOTH * (up + dn + lf + rt - 4.0f * c);
        }
        __syncthreads();
    }
    // result of step 33 (odd) landed in u0
    for (int p = threadIdx.x; p < NP; p += 256) {
        float v = u0[p];
        ws[OFF_AINV + b * NP + p] = 1.0f / v;
        u1[p] = logf(fmaxf(v, 1e-9f));
    }
    __syncthreads();
    for (int p = threadIdx.x; p < NP; p += 256) {
        int x = p >> 7, y = p & 127;
        float gx = (x == 0)   ? (u1[NN + y] - u1[y]) * INVH
                 : (x == 127) ? (u1[127 * NN + y] - u1[126 * NN + y]) * INVH
                 : (u1[(x + 1) * NN + y] - u1[(x - 1) * NN + y]) * (0.5f * INVH);
        float gy = (y == 0)   ? (u1[x * NN + 1] - u1[x * NN]) * INVH
                 : (y == 127) ? (u1[x * NN + 127] - u1[x * NN + 126]) * INVH
                 : (u1[x * NN + y + 1] - u1[x * NN + y - 1]) * (0.5f * INVH);
        ws[OFF_DLX + b * NP + p] = gx;
        ws[OFF_DLY + b * NP + p] = gy;
    }
}

// =====================================================================
// sigma_gamma for theta1/theta2
// =====================================================================
__global__ void sigma_kernel(const float* __restrict__ t1,
                             const float* __restrict__ t2, float* ws)
{
    long i = (long)blockIdx.x * blockDim.x + threadIdx.x;
    const long tot = (long)MM * NP;
    if (i < tot) {
        float r = t1[i];
        ws[OFF_S1 + i] = -0.33333334f + 0.41666667f / (1.0f + expf(-r * (1.0f / 0.15f)));
    } else if (i < 2 * tot) {
        float r = t2[i - tot];
        ws[OFF_S2 + (i - tot)] = -0.33333334f + 0.41666667f / (1.0f + expf(-r * (1.0f / 0.15f)));
    }
}

// salpha[p] = sum_m alpha[m] * s1[m,p]   (fixed-order, deterministic)
__global__ void salpha_kernel(const float* __restrict__ alpha, float* ws)
{
    int p = blockIdx.x * blockDim.x + threadIdx.x;
    if (p >= NP) return;
    float acc = 0.f;
    for (int m = 0; m < MM; ++m) acc += alpha[m] * ws[OFF_S1 + m * NP + p];
    ws[OFF_SALPHA + p] = acc;
}

// =====================================================================
// solve #1: per (b,m): rhs0 -> poisson -> grad -> alpha*s2-weighted
// accumulation into accx/accy.  One workgroup per (b,m), 4 WMMA GEMMs.
// =====================================================================
__global__ __launch_bounds__(256)
void solve1_kernel(const float* __restrict__ alpha, float* ws)
{
    extern __shared__ float smem[];
    float* ldsS = smem;
    float* ldsA = smem + NN * PIT;
    float* ldsT = smem + 2 * NN * PIT;

    const int bm = blockIdx.x;
    const int b  = bm >> 7;          // / MM
    const int m  = bm & 127;         // % MM
    const float* s1m   = ws + OFF_S1 + (long)m * NP;
    const float* s2m   = ws + OFF_S2 + (long)m * NP;
    const float* ainvb = ws + OFF_AINV + b * NP;
    const float* Sg    = ws + OFF_SMAT;
    const float* Dinv  = ws + OFF_DINV;
    float* accx = ws + OFF_ACCX + b * NP;
    float* accy = ws + OFF_ACCY + b * NP;

    for (int p = threadIdx.x; p < NP; p += 256) {
        int r = p >> 7, c = p & 127;
        ldsS[r * PIT + c] = Sg[p];
        float v = 0.f;
        if (r < 126 && c < 126) {
            int q = (r + 1) * NN + (c + 1);
            v = s1m[q] * ainvb[q];           // rhs0 interior
        }
        ldsA[r * PIT + c] = v;
    }
    __syncthreads();

    gemm128<0>(ldsS, ldsA, ldsT, nullptr, nullptr);   // T = S * r0
    __syncthreads();
    gemm128<1>(ldsT, ldsS, ldsA, Dinv, nullptr);      // A = (T*S) .* Dinv
    __syncthreads();
    gemm128<0>(ldsS, ldsA, ldsT, nullptr, nullptr);   // T = S * E
    __syncthreads();
    if (threadIdx.x < NN) {                           // zero p0 border
        ldsA[threadIdx.x] = 0.f;
        ldsA[threadIdx.x * PIT] = 0.f;
    }
    gemm128<2>(ldsT, ldsS, ldsA, nullptr, nullptr);   // ldsA = padded p0
    __syncthreads();

    const float al = alpha[m];
    for (int p = threadIdx.x; p < NP; p += 256) {
        int x = p >> 7, y = p & 127;
        float gx = (x == 0)   ? (ldsA[PIT + y] - ldsA[y]) * INVH
                 : (x == 127) ? (ldsA[127 * PIT + y] - ldsA[126 * PIT + y]) * INVH
                 : (ldsA[(x + 1) * PIT + y] - ldsA[(x - 1) * PIT + y]) * (0.5f * INVH);
        float gy = (y == 0)   ? (ldsA[x * PIT + 1] - ldsA[x * PIT]) * INVH
                 : (y == 127) ? (ldsA[x * PIT + 127] - ldsA[x * PIT + 126]) * INVH
                 : (ldsA[x * PIT + y + 1] - ldsA[x * PIT + y - 1]) * (0.5f * INVH);
        float w = al * s2m[p];
        atomicAdd(&accx[p], w * gx);
        atomicAdd(&accy[p], w * gy);
    }
}

// rhs1_sum[b] = a_inv*salpha + dlx*accx + dly*accy
__global__ void combine_kernel(float* ws)
{
    int i = blockIdx.x * blockDim.x + threadIdx.x;   // BB*NP threads
    if (i >= BB * NP) return;
    int p = i & (NP - 1);
    ws[OFF_RHS1 + i] = ws[OFF_AINV + i] * ws[OFF_SALPHA + p]
                     + ws[OFF_DLX + i] * ws[OFF_ACCX + i]
                     + ws[OFF_DLY + i] * ws[OFF_ACCY + i];
}

// =====================================================================
// solve #2: per b: poisson(rhs1_sum) -> d_out  (linearity collapses m)
// =====================================================================
__global__ __launch_bounds__(256)
void solve2_kernel(float* ws, float* __restrict__ out)
{
    extern __shared__ float smem[];
    float* ldsS = smem;
    float* ldsA = smem + NN * PIT;
    float* ldsT = smem + 2 * NN * PIT;

    const int b = blockIdx.x;
    const float* rhs = ws + OFF_RHS1 + b * NP;
    float* outb = out + b * NP;

    for (int p = threadIdx.x; p < NP; p += 256) {
        int r = p >> 7, c = p & 127;
        ldsS[r * PIT + c] = ws[OFF_SMAT + p];
        float v = 0.f;
        if (r < 126 && c < 126) v = rhs[(r + 1) * NN + (c + 1)];
        ldsA[r * PIT + c] = v;
    }
    __syncthreads();

    gemm128<0>(ldsS, ldsA, ldsT, nullptr, nullptr);
    __syncthreads();
    gemm128<1>(ldsT, ldsS, ldsA, ws + OFF_DINV, nullptr);
    __syncthreads();
    gemm128<0>(ldsS, ldsA, ldsT, nullptr, nullptr);
    __syncthreads();
    if (threadIdx.x < NN) {                 // zero output border
        outb[threadIdx.x] = 0.f;
        outb[threadIdx.x * NN] = 0.f;
    }
    gemm128<3>(ldsT, ldsS, nullptr, nullptr, outb);
}

// =====================================================================
extern "C" void kernel_launch(void* const* d_in, const int* in_sizes, int n_in,
                              void* d_out, int out_size, void* d_ws, size_t ws_size,
                              hipStream_t stream)
{
    const float* a     = (const float*)d_in[0];
    const float* th1   = (const float*)d_in[1];
    const float* th2   = (const float*)d_in[2];
    const float* alpha = (const float*)d_in[3];
    float* out = (float*)d_out;
    float* ws  = (float*)d_ws;

    const size_t SOLVE_SMEM = 3 * NN * PIT * sizeof(float);   // ~195 KB
    const size_t SMOOTH_SMEM = 2 * NP * sizeof(float);        // 128 KB

    init_kernel<<<(2 * BB * NP + 255) / 256, 256, 0, stream>>>(ws);
    smooth_kernel<<<BB, 256, SMOOTH_SMEM, stream>>>(a, ws);
    sigma_kernel<<<(2 * MM * NP + 255) / 256, 256, 0, stream>>>(th1, th2, ws);
    salpha_kernel<<<NP / 256, 256, 0, stream>>>(alpha, ws);
    solve1_kernel<<<BB * MM, 256, SOLVE_SMEM, stream>>>(alpha, ws);
    combine_kernel<<<(BB * NP + 255) / 256, 256, 0, stream>>>(ws);
    solve2_kernel<<<BB, 256, SOLVE_SMEM, stream>>>(ws, out);
}